// InstanceHead_23381801959899
// MI455X (gfx1250) — compile-verified
//
#include <hip/hip_runtime.h>
#include <hip/hip_bf16.h>

// ---------------- problem constants ----------------
#define NVOX    500000
#define LATENT  128
#define DESC    32
#define KSEG    256
#define OUTC    257            // K+1 output columns
#define BASE    130
#define BASE2   16900
#define TAU     0.1f
#define MINSC   0.5f
#define NEGF    (-1e30f)
#define NTILES  (NVOX / 16)    // 31250
#define GRID_CELLS 4400000     // max key ~4.377M + dkey margin
#define NBINS   16384
#define SELCAP  1024

typedef float v2f __attribute__((ext_vector_type(2)));
typedef float v8f __attribute__((ext_vector_type(8)));

// ---------------- helpers ----------------
__device__ __forceinline__ int voxel_key(const int* __restrict__ coords, int i) {
    const int* c = coords + 4 * i;
    return (c[0] * BASE + c[1] + 1) * BASE2 + (c[2] + 1) * BASE + (c[3] + 1);
}

// 7-point stencil key offsets: dx^2+dy^2+dz^2 <= 1.21 -> center + 6 faces
__device__ __forceinline__ void stencil(int* dk) {
    dk[0] = 0; dk[1] = BASE2; dk[2] = -BASE2; dk[3] = BASE; dk[4] = -BASE; dk[5] = 1; dk[6] = -1;
}

__device__ __forceinline__ bool eval_peak(int i, const int* __restrict__ coords,
                                          const float* __restrict__ scores,
                                          const int* __restrict__ grid, float& s_out) {
    float s = scores[i];
    s_out = s;
    if (!(s > TAU)) return false;
    int key = voxel_key(coords, i);
    int dk[7]; stencil(dk);
    float nmax = NEGF;
#pragma unroll
    for (int d = 0; d < 7; ++d) {
        int g = grid[key + dk[d]];
        if (g > 0) {
            float sj = scores[g - 1];
            if (sj > TAU && sj > nmax) nmax = sj;
        }
    }
    return (s >= nmax - 1e-6f) && (s >= MINSC);
}

// ---------------- 1: init workspace ----------------
__global__ void ih_init(int* __restrict__ grid, int* __restrict__ ownerInit,
                        int* __restrict__ hist, int* __restrict__ ownerCount,
                        int* __restrict__ ctr) {
    int i = blockIdx.x * blockDim.x + threadIdx.x;
    if (i < GRID_CELLS) grid[i] = 0;
    if (i < NVOX)       ownerInit[i] = -1;
    if (i < NBINS)      hist[i] = 0;
    if (i < KSEG)       ownerCount[i] = 0;
    if (i < 16)         ctr[i] = 0;
}

// ---------------- 2: build dense key grid ----------------
__global__ void ih_build_grid(const int* __restrict__ coords, int* __restrict__ grid) {
    int i = blockIdx.x * blockDim.x + threadIdx.x;
    if (i >= NVOX) return;
    grid[voxel_key(coords, i)] = i + 1;
}

// ---------------- 3: peak detection -> score histogram ----------------
__global__ void ih_peak_hist(const int* __restrict__ coords, const float* __restrict__ scores,
                             const int* __restrict__ grid, int* __restrict__ hist) {
    int i = blockIdx.x * blockDim.x + threadIdx.x;
    if (i >= NVOX) return;
    float s;
    if (eval_peak(i, coords, scores, grid, s)) {
        int b = (int)(s * (float)NBINS);
        b = b < 0 ? 0 : (b > NBINS - 1 ? NBINS - 1 : b);
        atomicAdd(&hist[b], 1);
    }
}

// ---------------- 4: find threshold bin for top-K ----------------
__global__ void ih_threshold(const int* __restrict__ hist, int* __restrict__ ctr) {
    if (threadIdx.x != 0) return;
    int cum = 0, T = 0;
    for (int b = NBINS - 1; b >= 0; --b) {
        cum += hist[b];
        if (cum >= KSEG) { T = b; break; }
    }
    ctr[1] = T;
}

// ---------------- 5: collect candidates with bin >= T (deterministic set) ----------------
__global__ void ih_select(const int* __restrict__ coords, const float* __restrict__ scores,
                          const int* __restrict__ grid, const int* __restrict__ ctr_in,
                          int* __restrict__ ctr, float* __restrict__ sel_s, int* __restrict__ sel_i) {
    int i = blockIdx.x * blockDim.x + threadIdx.x;
    if (i >= NVOX) return;
    float s;
    if (!eval_peak(i, coords, scores, grid, s)) return;
    int b = (int)(s * (float)NBINS);
    b = b < 0 ? 0 : (b > NBINS - 1 ? NBINS - 1 : b);
    if (b >= ctr_in[1]) {
        int pos = atomicAdd(&ctr[0], 1);
        if (pos < SELCAP) { sel_s[pos] = s; sel_i[pos] = i; }
    }
}

// ---------------- 6: exact rank (score desc, idx asc), write conf + ownerInit ----------------
__global__ __launch_bounds__(256) void ih_rank(const float* __restrict__ sel_s,
                                               const int* __restrict__ sel_i,
                                               const int* __restrict__ ctr,
                                               float* __restrict__ conf,    // d_out[0..255]
                                               int* __restrict__ ownerInit) {
    int n = ctr[0]; if (n > SELCAP) n = SELCAP;
    int t = threadIdx.x;
    conf[t] = 0.0f;
    __syncthreads();
    for (int e = t; e < n; e += 256) {
        float s = sel_s[e]; int id = sel_i[e];
        int rank = 0;
        for (int j = 0; j < n; ++j) {
            float sj = sel_s[j];
            rank += (sj > s) || (sj == s && sel_i[j] < id);
        }
        if (rank < KSEG) { conf[rank] = s; ownerInit[id] = rank; }
    }
}

// ---------------- 7: owner per voxel (max peak-rank in stencil) + counts ----------------
__global__ void ih_owner(const int* __restrict__ coords, const float* __restrict__ scores,
                         const int* __restrict__ grid, const int* __restrict__ ownerInit,
                         int* __restrict__ owner, int* __restrict__ ownerCount) {
    int i = blockIdx.x * blockDim.x + threadIdx.x;
    if (i >= NVOX) return;
    int ow = -1;
    float s = scores[i];
    if (s > TAU) {
        int key = voxel_key(coords, i);
        int dk[7]; stencil(dk);
#pragma unroll
        for (int d = 0; d < 7; ++d) {
            int g = grid[key + dk[d]];
            if (g > 0) {
                int j = g - 1;
                if (scores[j] > TAU) {
                    int oi = ownerInit[j];
                    if (oi > ow) ow = oi;
                }
            }
        }
    }
    owner[i] = ow;
    if (ow >= 0) atomicAdd(&ownerCount[ow], 1);
}

// ---------------- 8: serial prefix over 256 owners ----------------
__global__ void ih_prefix(const int* __restrict__ ownerCount, int* __restrict__ ownerStart,
                          int* __restrict__ ownerFill) {
    if (threadIdx.x != 0) return;
    int acc = 0;
    for (int k = 0; k < KSEG; ++k) { ownerStart[k] = acc; ownerFill[k] = acc; acc += ownerCount[k]; }
}

// ---------------- 9: scatter voxel ids into per-owner lists ----------------
__global__ void ih_scatter(const int* __restrict__ owner, int* __restrict__ ownerFill,
                           int* __restrict__ ownedIdx) {
    int i = blockIdx.x * blockDim.x + threadIdx.x;
    if (i >= NVOX) return;
    int ow = owner[i];
    if (ow >= 0) {
        int pos = atomicAdd(&ownerFill[ow], 1);
        ownedIdx[pos] = i;
    }
}

// ---------------- 10: per-owner coalesced feature mean (no atomics) ----------------
__global__ __launch_bounds__(128) void ih_segmean(const float* __restrict__ feats,
                                                  const int* __restrict__ ownedIdx,
                                                  const int* __restrict__ ownerStart,
                                                  const int* __restrict__ ownerCount,
                                                  float* __restrict__ feat_mean) {
    int o = blockIdx.x;          // 256 blocks
    int f = threadIdx.x;         // 128 features
    int s = ownerStart[o], c = ownerCount[o];
    float acc = 0.0f;
    for (int i = 0; i < c; ++i)
        acc += feats[(size_t)ownedIdx[s + i] * LATENT + f];
    feat_mean[o * LATENT + f] = acc / fmaxf((float)c, 1.0f);
}

// ---------------- 11: cols = [bg ; conf * (feat_mean@Wc + bc)]  (257x32) ----------------
__global__ void ih_cols(const float* __restrict__ feat_mean, const float* __restrict__ Wc,
                        const float* __restrict__ bc, const float* __restrict__ bg,
                        const float* __restrict__ conf, float* __restrict__ cols) {
    int idx = blockIdx.x * blockDim.x + threadIdx.x;
    if (idx >= OUTC * DESC) return;
    int row = idx / DESC, d = idx % DESC;
    if (row == 0) { cols[idx] = bg[d]; return; }
    int k = row - 1;
    float s = 0.0f;
    for (int t = 0; t < LATENT; ++t)
        s += feat_mean[k * LATENT + t] * Wc[t * DESC + d];
    cols[idx] = conf[k] * (s + bc[d]);
}

// ---------------- 12: fused tile GEMM with V_WMMA_F32_16X16X4_F32 ----------------
// per wave: one 16-row tile
//   stage1: vd(16x32) = feats(16x128) @ Wv(128x32) + bv   (64 WMMA)
//   LDS round trip: C-layout -> row-major 16x32 -> A-layout
//   stage2: out(16x257) = vd(16x32) @ cols(257x32)^T      (17 N-tiles * 8 WMMA)
//
// Wv is staged into LDS *pre-swizzled into B-fragment order* so each lane's
// B fragment is one aligned ds_load_b64 (no cross-stride assembly / v_movs):
//   wv_lds[g*128 + n*4 + p*2 + j] = Wv[(g*4 + 2p + j)][n],  g=K/4, p=pair, j=0/1
__global__ __launch_bounds__(256) void ih_gemm(const float* __restrict__ feats,
                                               const float* __restrict__ Wv,
                                               const float* __restrict__ bv,
                                               const float* __restrict__ cols,
                                               float* __restrict__ out) {
    __shared__ float wv_lds[LATENT * DESC];     // 16 KB: Wv, B-fragment swizzled
    __shared__ float vd_lds[8 * 16 * DESC];     // 16 KB: per-wave voxel_desc tile

    int tid  = threadIdx.x;
    int wave = tid >> 5;
    int lane = tid & 31;

    int tile = blockIdx.x * 8 + wave;
    if (tile >= NTILES) tile = NTILES - 1;      // duplicate work, identical values
    int row0 = tile * 16;

    int m    = lane & 15;                        // M (stage1 A) / N (B,C)
    int koff = (lane < 16) ? 0 : 2;              // K sub-offset for A/B layout
    int mrow = (lane < 16) ? 0 : 8;              // C-layout row base

    const float* arow = feats + (size_t)(row0 + m) * LATENT;
    // prefetch this lane's A row (512 B) while we stage Wv through LDS
    __builtin_prefetch(arow, 0, 1);
    __builtin_prefetch(arow + 64, 0, 1);

    // stage Wv into LDS, swizzled into contiguous per-lane B fragments
    for (int t = tid; t < LATENT * DESC; t += 256) {
        int j = t & 1;
        int p = (t >> 1) & 1;
        int n = (t >> 2) & 31;
        int g = t >> 7;
        wv_lds[t] = Wv[(g * 4 + 2 * p + j) * DESC + n];
    }
    __syncthreads();

    // ---- stage 1 ----
    v8f c0, c1;
    float b0 = bv[m], b1 = bv[16 + m];
#pragma unroll
    for (int r = 0; r < 8; ++r) { c0[r] = b0; c1[r] = b1; }

    const float* wvp = wv_lds + m * 4 + (koff >> 1) * 2;   // + g*128 per K-group, +64 for n-tile 1
#pragma unroll 4
    for (int k0 = 0; k0 < LATENT; k0 += 4) {
        int g = k0 >> 2;
        v2f a  = *(const v2f*)(arow + k0 + koff);          // global b64: (M=m, K=k0+koff..+1)
        v2f bA = *(const v2f*)(wvp + g * 128);             // ds_load_b64
        v2f bB = *(const v2f*)(wvp + g * 128 + 64);        // ds_load_b64
        c0 = __builtin_amdgcn_wmma_f32_16x16x4_f32(false, a, false, bA, (short)0, c0, false, false);
        c1 = __builtin_amdgcn_wmma_f32_16x16x4_f32(false, a, false, bB, (short)0, c1, false, false);
    }

    // write vd tile to LDS row-major: element (M=mrow+r, N=m)
    float* vl = &vd_lds[wave * 16 * DESC];
#pragma unroll
    for (int r = 0; r < 8; ++r) {
        vl[(mrow + r) * DESC + m]      = c0[r];
        vl[(mrow + r) * DESC + 16 + m] = c1[r];
    }
    __syncthreads();

    // ---- stage 2 ----
    const float* vrow = vl + m * DESC;                          // row m of vd (A-matrix row)
    for (int n0 = 0; n0 < OUTC; n0 += 16) {                     // 17 N-tiles (last partial)
        v8f acc;
#pragma unroll
        for (int r = 0; r < 8; ++r) acc[r] = 0.0f;
        int n   = n0 + m;
        int ncl = (n < OUTC) ? n : (OUTC - 1);
        const float* colrow = cols + (size_t)ncl * DESC;        // B[k][n] = cols[n][k]
#pragma unroll
        for (int k0 = 0; k0 < DESC; k0 += 4) {
            v2f a = *(const v2f*)(vrow + k0 + koff);            // ds_load_b64
            v2f b = *(const v2f*)(colrow + k0 + koff);          // global b64, L2-resident
            acc = __builtin_amdgcn_wmma_f32_16x16x4_f32(false, a, false, b, (short)0, acc, false, false);
        }
        if (n < OUTC) {
            float* op = out + (size_t)(row0 + mrow) * OUTC + n;
#pragma unroll
            for (int r = 0; r < 8; ++r) op[(size_t)r * OUTC] = acc[r];
        }
    }
}

// ---------------- host launch ----------------
extern "C" void kernel_launch(void* const* d_in, const int* in_sizes, int n_in,
                              void* d_out, int out_size, void* d_ws, size_t ws_size,
                              hipStream_t stream) {
    const int*   coords = (const int*)  d_in[0];
    const float* feats  = (const float*)d_in[1];
    const float* scores = (const float*)d_in[2];   // (N,1) contiguous
    const float* Wv     = (const float*)d_in[3];
    const float* bv     = (const float*)d_in[4];
    const float* Wc     = (const float*)d_in[5];
    const float* bc     = (const float*)d_in[6];
    const float* bg     = (const float*)d_in[7];

    float* conf = (float*)d_out;                   // 256 floats
    float* out  = (float*)d_out + KSEG;            // 500000 x 257

    // workspace carve-up (256B aligned)
    char*  base = (char*)d_ws;
    size_t off  = 0;
    auto carve = [&](size_t bytes) { char* p = base + off; off = (off + bytes + 255) & ~(size_t)255; return p; };
    int*   grid       = (int*)  carve((size_t)GRID_CELLS * 4);
    int*   ownerInit  = (int*)  carve((size_t)NVOX * 4);
    int*   owner      = (int*)  carve((size_t)NVOX * 4);
    int*   ownedIdx   = (int*)  carve((size_t)NVOX * 4);
    int*   hist       = (int*)  carve((size_t)NBINS * 4);
    int*   ownerCount = (int*)  carve(KSEG * 4);
    int*   ownerStart = (int*)  carve(KSEG * 4);
    int*   ownerFill  = (int*)  carve(KSEG * 4);
    int*   ctr        = (int*)  carve(16 * 4);
    float* sel_s      = (float*)carve(SELCAP * 4);
    int*   sel_i      = (int*)  carve(SELCAP * 4);
    float* feat_mean  = (float*)carve((size_t)KSEG * LATENT * 4);
    float* cols       = (float*)carve((size_t)OUTC * DESC * 4);
    (void)ws_size; (void)in_sizes; (void)n_in; (void)out_size;

    const int TB = 256;
    dim3 blkInit((GRID_CELLS + TB - 1) / TB);
    dim3 blkN((NVOX + TB - 1) / TB);

    ih_init      <<<blkInit, TB, 0, stream>>>(grid, ownerInit, hist, ownerCount, ctr);
    ih_build_grid<<<blkN,    TB, 0, stream>>>(coords, grid);
    ih_peak_hist <<<blkN,    TB, 0, stream>>>(coords, scores, grid, hist);
    ih_threshold <<<1,       32, 0, stream>>>(hist, ctr);
    ih_select    <<<blkN,    TB, 0, stream>>>(coords, scores, grid, ctr, ctr, sel_s, sel_i);
    ih_rank      <<<1,       TB, 0, stream>>>(sel_s, sel_i, ctr, conf, ownerInit);
    ih_owner     <<<blkN,    TB, 0, stream>>>(coords, scores, grid, ownerInit, owner, ownerCount);
    ih_prefix    <<<1,       32, 0, stream>>>(ownerCount, ownerStart, ownerFill);
    ih_scatter   <<<blkN,    TB, 0, stream>>>(owner, ownerFill, ownedIdx);
    ih_segmean   <<<KSEG,   128, 0, stream>>>(feats, ownedIdx, ownerStart, ownerCount, feat_mean);
    ih_cols      <<<(OUTC * DESC + TB - 1) / TB, TB, 0, stream>>>(feat_mean, Wc, bc, bg, conf, cols);
    ih_gemm      <<<(NTILES + 7) / 8, TB, 0, stream>>>(feats, Wv, bv, cols, out);
}